// EmbeddingHead_35287451304068
// MI455X (gfx1250) — compile-verified
//
#include <hip/hip_runtime.h>
#include <hip/hip_bf16.h>
#include <math.h>

// ---------------------------------------------------------------------------
// EmbeddingHead for MI455X (gfx1250, wave32).
//   Stage 1: stream 308MB of frames once (bandwidth bound, ~13us floor)
//            computing SPP (90) + ROI (147) features per frame.
//   Stage 2/3: tiny f32 GEMMs via V_WMMA_F32_16X16X4_F32 (matches f32 ref).
// ---------------------------------------------------------------------------

typedef float v2f __attribute__((ext_vector_type(2)));
typedef float v8f __attribute__((ext_vector_type(8)));

#define NFRAME 512
#define IMG_H 224
#define IMG_W 224
#define NPIX (IMG_H * IMG_W)
#define KSPP 92    // 90 padded to mult of 4
#define KROI 148   // 147 padded to mult of 4
#define EMB 2048
#define HID 4096
#define NOUT 64
#define NEGKEY 0x007FFFFFu   // monotone key of -inf

// order-preserving float->u32 key so we can use integer LDS atomicMax
__device__ __forceinline__ unsigned fkey(float f) {
  unsigned b = __float_as_uint(f);
  return (b & 0x80000000u) ? ~b : (b | 0x80000000u);
}
__device__ __forceinline__ float funkey(unsigned u) {
  unsigned b = (u & 0x80000000u) ? (u & 0x7FFFFFFFu) : ~u;
  return __uint_as_float(b);
}

// ---------------------------------------------------------------------------
// Pooling kernel: one workgroup per (frame, channel).  Thread t owns image
// row t (t<224); scans 56 float4 chunks keeping per-row column-segment maxes
// in registers (loop fully unrolled -> constant indices), then reduces across
// rows with LDS u32 atomic max.
// ---------------------------------------------------------------------------
__global__ __launch_bounds__(256) void pool_kernel(
    const float* __restrict__ frames, const float* __restrict__ boxes,
    float* __restrict__ spp_feat, float* __restrict__ roi_feat) {
  __shared__ unsigned s_l4[16], s_l3[9], s_l2[4], s_l1[1], s_roi[49];
  const int wg = blockIdx.x;
  const int n = wg / 3, ch = wg % 3;
  const int tid = threadIdx.x;

  if (tid < 16) s_l4[tid] = NEGKEY;
  if (tid < 9)  s_l3[tid] = NEGKEY;
  if (tid < 4)  s_l2[tid] = NEGKEY;
  if (tid == 0) s_l1[0]   = NEGKEY;
  if (tid < 49) s_roi[tid] = NEGKEY;

  // ROI box (uniform across the workgroup). jnp.round == rintf (RTE).
  const int x1 = (int)rintf(boxes[n * 4 + 0]);
  const int y1 = (int)rintf(boxes[n * 4 + 1]);
  const int x2 = (int)rintf(boxes[n * 4 + 2]);
  const int y2 = (int)rintf(boxes[n * 4 + 3]);
  const float roi_w = (float)max(x2 - x1 + 1, 1);
  const float roi_h = (float)max(y2 - y1 + 1, 1);
  int ws[7], we[7], hs[7], he[7];
#pragma unroll
  for (int p = 0; p < 7; ++p) {
    hs[p] = min(max((int)floorf((float)p * roi_h / 7.0f) + y1, 0), IMG_H);
    he[p] = min(max((int)ceilf((float)(p + 1) * roi_h / 7.0f) + y1, 0), IMG_H);
    ws[p] = min(max((int)floorf((float)p * roi_w / 7.0f) + x1, 0), IMG_W);
    we[p] = min(max((int)ceilf((float)(p + 1) * roi_w / 7.0f) + x1, 0), IMG_W);
  }
  __syncthreads();

  const int r = tid;
  if (r < IMG_H) {
    const float4* row =
        (const float4*)(frames + (size_t)(n * 3 + ch) * NPIX + r * IMG_W);
    float l4c[4], l3c[3], l2c[2], roic[7];
    float l1c = -INFINITY;
#pragma unroll
    for (int j = 0; j < 4; ++j) l4c[j] = -INFINITY;
#pragma unroll
    for (int j = 0; j < 3; ++j) l3c[j] = -INFINITY;
    l2c[0] = l2c[1] = -INFINITY;
#pragma unroll
    for (int j = 0; j < 7; ++j) roic[j] = -INFINITY;

#pragma unroll
    for (int i = 0; i < 56; ++i) {
      const float4 q = row[i];
      const int c0 = 4 * i;
      const float m = fmaxf(fmaxf(q.x, q.y), fmaxf(q.z, q.w));
      l4c[i / 14] = fmaxf(l4c[i / 14], m);   // 56-wide bins, chunk-aligned
      l2c[i / 28] = fmaxf(l2c[i / 28], m);   // 112-wide bins
      l1c = fmaxf(l1c, m);
      // L=3 col bins: [0,75) [74,150) [149,224)  (1-px overlaps)
      if (i <= 17) {
        l3c[0] = fmaxf(l3c[0], m);
      } else if (i == 18) {  // cols 72,73,74,75
        l3c[0] = fmaxf(l3c[0], fmaxf(fmaxf(q.x, q.y), q.z));
        l3c[1] = fmaxf(l3c[1], fmaxf(q.z, q.w));
      } else if (i <= 36) {
        l3c[1] = fmaxf(l3c[1], m);
      } else if (i == 37) {  // cols 148,149,150,151
        l3c[1] = fmaxf(l3c[1], fmaxf(q.x, q.y));
        l3c[2] = fmaxf(l3c[2], fmaxf(fmaxf(q.y, q.z), q.w));
      } else {
        l3c[2] = fmaxf(l3c[2], m);
      }
      // ROI column bins: ws/we are uniform -> scalar branches
#pragma unroll
      for (int pw = 0; pw < 7; ++pw) {
        const int a = ws[pw], b = we[pw];
        if (a <= c0 && c0 + 4 <= b) {
          roic[pw] = fmaxf(roic[pw], m);
        } else if (c0 < b && a < c0 + 4) {
          if (c0 + 0 >= a && c0 + 0 < b) roic[pw] = fmaxf(roic[pw], q.x);
          if (c0 + 1 >= a && c0 + 1 < b) roic[pw] = fmaxf(roic[pw], q.y);
          if (c0 + 2 >= a && c0 + 2 < b) roic[pw] = fmaxf(roic[pw], q.z);
          if (c0 + 3 >= a && c0 + 3 < b) roic[pw] = fmaxf(roic[pw], q.w);
        }
      }
    }

    // cross-row reduction via LDS integer atomic max (ds_max_u32)
    const int rb4 = r / 56;
#pragma unroll
    for (int j = 0; j < 4; ++j) atomicMax(&s_l4[rb4 * 4 + j], fkey(l4c[j]));
    if (r <= 74) {
#pragma unroll
      for (int j = 0; j < 3; ++j) atomicMax(&s_l3[j], fkey(l3c[j]));
    }
    if (r >= 74 && r <= 149) {
#pragma unroll
      for (int j = 0; j < 3; ++j) atomicMax(&s_l3[3 + j], fkey(l3c[j]));
    }
    if (r >= 149) {
#pragma unroll
      for (int j = 0; j < 3; ++j) atomicMax(&s_l3[6 + j], fkey(l3c[j]));
    }
    const int rb2 = r / 112;
#pragma unroll
    for (int j = 0; j < 2; ++j) atomicMax(&s_l2[rb2 * 2 + j], fkey(l2c[j]));
    atomicMax(&s_l1[0], fkey(l1c));
#pragma unroll
    for (int ph = 0; ph < 7; ++ph) {
      if (r >= hs[ph] && r < he[ph]) {
#pragma unroll
        for (int pw = 0; pw < 7; ++pw)
          atomicMax(&s_roi[ph * 7 + pw], fkey(roic[pw]));
      }
    }
  }
  __syncthreads();

  // emit features (reference concat order: L4 | L3 | L2 | L1, channel-major
  // inside each level block); ROI: (ch, ph, pw); empty ROI bins -> 0
  float* srow = spp_feat + (size_t)n * KSPP;
  float* rrow = roi_feat + (size_t)n * KROI;
  if (tid < 16) {
    srow[ch * 16 + tid] = funkey(s_l4[tid]);
  } else if (tid < 25) {
    srow[48 + ch * 9 + (tid - 16)] = funkey(s_l3[tid - 16]);
  } else if (tid < 29) {
    srow[75 + ch * 4 + (tid - 25)] = funkey(s_l2[tid - 25]);
  } else if (tid == 29) {
    srow[87 + ch] = funkey(s_l1[0]);
  }
  if (tid >= 32 && tid < 81) {
    const unsigned u = s_roi[tid - 32];
    rrow[ch * 49 + (tid - 32)] = (u == NEGKEY) ? 0.0f : funkey(u);
  }
  if (ch == 0 && tid == 81) {  // K-dim zero padding
    srow[90] = 0.0f;
    srow[91] = 0.0f;
    rrow[147] = 0.0f;
  }
}

// ---------------------------------------------------------------------------
// Zero-pad weight copies so the WMMA K loop is uniform.
// ---------------------------------------------------------------------------
__global__ __launch_bounds__(256) void pad_weights(
    const float* __restrict__ Wspp, const float* __restrict__ Wroi,
    float* __restrict__ WsppP, float* __restrict__ WroiP) {
  const int idx = blockIdx.x * blockDim.x + threadIdx.x;
  const int t1 = EMB * KSPP;
  const int t2 = EMB * KROI;
  if (idx < t1) {
    const int j = idx / KSPP, k = idx % KSPP;
    WsppP[idx] = (k < 90) ? Wspp[j * 90 + k] : 0.0f;
  } else if (idx < t1 + t2) {
    const int t = idx - t1;
    const int j = t / KROI, k = t % KROI;
    WroiP[t] = (k < 147) ? Wroi[j * 147 + k] : 0.0f;
  }
}

// D = A x B + C, f32, 16x16 tile, K=4 per issue.
__device__ __forceinline__ v8f wmma4(v2f a, v2f b, v8f c) {
  return __builtin_amdgcn_wmma_f32_16x16x4_f32(false, a, false, b, (short)0, c,
                                               false, false);
}

// ---------------------------------------------------------------------------
// GEMM1+2 fused: hidden[512][4096] = relu(feat @ W^T + bias) for both halves.
// One wave per 16x16 output tile.  8192 tiles total.
// ---------------------------------------------------------------------------
__global__ __launch_bounds__(256) void gemm_hidden(
    const float* __restrict__ feat_spp, const float* __restrict__ feat_roi,
    const float* __restrict__ WsppP, const float* __restrict__ WroiP,
    const float* __restrict__ b_spp, const float* __restrict__ b_roi,
    float* __restrict__ H) {
  const int wave = (blockIdx.x * blockDim.x + threadIdx.x) >> 5;
  const int lane = threadIdx.x & 31;
  const int mt = wave & 31;   // 32 M tiles of 16 frames
  const int ntt = wave >> 5;  // 256 N tiles of 16 cols (128 spp + 128 roi)

  const float* feat;
  const float* W;
  const float* bias;
  int Kp, colbase, ncl;
  if (ntt < 128) {
    feat = feat_spp; W = WsppP; bias = b_spp; Kp = KSPP;
    colbase = 0; ncl = ntt * 16;
  } else {
    feat = feat_roi; W = WroiP; bias = b_roi; Kp = KROI;
    colbase = EMB; ncl = (ntt - 128) * 16;
  }
  const int rr = lane & 15, hi = lane >> 4;
  const float* ap = feat + (size_t)(mt * 16 + rr) * Kp + 2 * hi;  // A: 16xK
  const float* bp = W + (size_t)(ncl + rr) * Kp + 2 * hi;         // B: Kx16
  v8f acc = {};
  for (int k0 = 0; k0 < Kp; k0 += 4) {
    const v2f a = *(const v2f*)(ap + k0);
    const v2f b = *(const v2f*)(bp + k0);
    acc = wmma4(a, b, acc);
  }
  const float bv = bias[ncl + rr];
  const int col = colbase + ncl + rr;
#pragma unroll
  for (int i = 0; i < 8; ++i) {
    const float v = fmaxf(acc[i] + bv, 0.0f);
    H[(size_t)(mt * 16 + hi * 8 + i) * HID + col] = v;
  }
}

// ---------------------------------------------------------------------------
// GEMM3: out[512][64] = relu(H @ W_fc7^T + b_fc7).  128 tiles, K=4096.
// ---------------------------------------------------------------------------
__global__ __launch_bounds__(256) void gemm_out(
    const float* __restrict__ H, const float* __restrict__ Wfc7,
    const float* __restrict__ bfc7, float* __restrict__ out) {
  const int wave = (blockIdx.x * blockDim.x + threadIdx.x) >> 5;
  const int lane = threadIdx.x & 31;
  const int mt = wave & 31;   // 32 M tiles
  const int nt = wave >> 5;   // 4 N tiles
  const int rr = lane & 15, hi = lane >> 4;
  const float* ap = H + (size_t)(mt * 16 + rr) * HID + 2 * hi;
  const float* bp = Wfc7 + (size_t)(nt * 16 + rr) * HID + 2 * hi;
  v8f acc = {};
  for (int k0 = 0; k0 < HID; k0 += 4) {
    const v2f a = *(const v2f*)(ap + k0);
    const v2f b = *(const v2f*)(bp + k0);
    acc = wmma4(a, b, acc);
  }
  const float bv = bfc7[nt * 16 + rr];
#pragma unroll
  for (int i = 0; i < 8; ++i) {
    const float v = fmaxf(acc[i] + bv, 0.0f);
    out[(size_t)(mt * 16 + hi * 8 + i) * NOUT + nt * 16 + rr] = v;
  }
}

// ---------------------------------------------------------------------------
extern "C" void kernel_launch(void* const* d_in, const int* in_sizes, int n_in,
                              void* d_out, int out_size, void* d_ws,
                              size_t ws_size, hipStream_t stream) {
  const float* frames = (const float*)d_in[0];
  const float* boxes  = (const float*)d_in[1];
  const float* W_spp  = (const float*)d_in[2];
  const float* b_spp  = (const float*)d_in[3];
  const float* W_roi  = (const float*)d_in[4];
  const float* b_roi  = (const float*)d_in[5];
  const float* W_fc7  = (const float*)d_in[6];
  const float* b_fc7  = (const float*)d_in[7];
  float* out = (float*)d_out;

  float* ws = (float*)d_ws;
  float* spp_feat = ws;                          // 512*92
  float* roi_feat = spp_feat + NFRAME * KSPP;    // 512*148
  float* WsppP    = roi_feat + NFRAME * KROI;    // 2048*92
  float* WroiP    = WsppP + EMB * KSPP;          // 2048*148
  float* H        = WroiP + EMB * KROI;          // 512*4096  (~10.9MB total)

  const int padN = EMB * KSPP + EMB * KROI;
  pad_weights<<<(padN + 255) / 256, 256, 0, stream>>>(W_spp, W_roi, WsppP,
                                                      WroiP);
  pool_kernel<<<NFRAME * 3, 256, 0, stream>>>(frames, boxes, spp_feat,
                                              roi_feat);
  gemm_hidden<<<1024, 256, 0, stream>>>(spp_feat, roi_feat, WsppP, WroiP,
                                        b_spp, b_roi, H);
  gemm_out<<<16, 256, 0, stream>>>(H, W_fc7, b_fc7, out);
}